// GATwithEdgeAttr_50903952392628
// MI455X (gfx1250) — compile-verified
//
#include <hip/hip_runtime.h>
#include <math.h>

// Problem constants (from reference)
#define NNODES 50000
#define NEDGES 800000
#define DIM    128        // node feature dim (== C)
#define NH     2          // heads
#define CH     128        // channels per head
#define EDIM   8          // edge attr dim
#define HC     (NH * CH)  // 256
#define LEAKY  0.2f
#define SMEPS  1e-16f
#define LNEPS  1e-5f

typedef __attribute__((ext_vector_type(2))) float v2f;
typedef __attribute__((ext_vector_type(8))) float v8f;

__device__ __forceinline__ float atomAddF32(float* p, float v) {
    return __hip_atomic_fetch_add(p, v, __ATOMIC_RELAXED, __HIP_MEMORY_SCOPE_AGENT);
}

// float atomic max via int/uint punning (dst initialized to -inf)
__device__ __forceinline__ void atomMaxF32(float* addr, float v) {
    if (v >= 0.0f) atomicMax((int*)addr, __float_as_int(v));
    else           atomicMin((unsigned int*)addr, __float_as_uint(v));
}

// ---------------------------------------------------------------------------
// Kernel 0: fold the edge path.
//   we[d][h]  = sum_c W_le[d, h*CH+c] * att_edge[h,c]        (DIM x NH)
//   M[k][h]   = sum_d W_ep[k,d] * we[d][h]                   (EDIM x NH)
//   c0[h]     = sum_d b_ep[d]   * we[d][h]
// One 256-thread block.
// ---------------------------------------------------------------------------
__global__ void prep_fold_kernel(const float* __restrict__ W_le,
                                 const float* __restrict__ att_edge,
                                 const float* __restrict__ W_ep,
                                 const float* __restrict__ b_ep,
                                 float* __restrict__ Mfold,   // [EDIM*NH]
                                 float* __restrict__ c0)      // [NH]
{
    __shared__ float we_s[DIM * NH];
    int t = threadIdx.x;            // 0..255
    int d = t >> 1, hh = t & 1;
    const float* wrow = W_le + d * HC + hh * CH;
    const float* arow = att_edge + hh * CH;
    float s = 0.f;
    for (int c = 0; c < CH; ++c) s += wrow[c] * arow[c];
    we_s[d * NH + hh] = s;
    __syncthreads();
    if (t < EDIM * NH) {
        int k = t >> 1; int h2 = t & 1;
        float acc = 0.f;
        for (int dd = 0; dd < DIM; ++dd) acc += W_ep[k * DIM + dd] * we_s[dd * NH + h2];
        Mfold[t] = acc;             // layout [k][h] = k*NH+h
    }
    if (t < NH) {
        float acc = 0.f;
        for (int dd = 0; dd < DIM; ++dd) acc += b_ep[dd] * we_s[dd * NH + t];
        c0[t] = acc;
    }
}

// ---------------------------------------------------------------------------
// Kernel 1: init accumulators (smax = -inf, denom = 0, ypre = 0)
// ---------------------------------------------------------------------------
__global__ void init_kernel(float* __restrict__ smax,
                            float* __restrict__ denom,
                            float* __restrict__ ypre)
{
    int i = blockIdx.x * blockDim.x + threadIdx.x;
    if (i < NNODES * NH) { smax[i] = -__builtin_inff(); denom[i] = 0.f; }
    int stride = gridDim.x * blockDim.x;
    for (int j = i; j < NNODES * DIM; j += stride) ypre[j] = 0.f;
}

// ---------------------------------------------------------------------------
// Kernel 2: h = x @ W_lin  via V_WMMA_F32_16X16X4_F32 (full f32 precision).
// Grid (3125, 2), 256 threads = 8 waves; wave w handles col-tile blockIdx.y*8+w.
// A 16x4 frag: lane(0..15)=M, VGPR0:K=2*half, VGPR1:K=2*half+1 (half=lane>>4).
// B 4x16 frag: mirrored (row striped across lanes within VGPR).
// C/D 16x16:  VGPR j -> (M = j + 8*half, N = lane&15).
// ---------------------------------------------------------------------------
__global__ __launch_bounds__(256)
void node_proj_wmma_kernel(const float* __restrict__ X,     // [NNODES][DIM]
                           const float* __restrict__ W,     // [DIM][HC]
                           float* __restrict__ Hbuf)        // [NNODES][HC]
{
    const int wave   = threadIdx.x >> 5;
    const int lane   = threadIdx.x & 31;
    const int half   = lane >> 4;
    const int lanelo = lane & 15;
    const int row0   = blockIdx.x * 16;                 // 50000/16 = 3125 exact
    const int col    = (blockIdx.y * 8 + wave) * 16 + lanelo;

    v8f acc = {};
    const float* xrow = X + (row0 + lanelo) * DIM;
    #pragma unroll 4
    for (int k0 = 0; k0 < DIM; k0 += 4) {
        const int ka = k0 + 2 * half;
        v2f a, b;
        a.x = xrow[ka];
        a.y = xrow[ka + 1];
        b.x = W[ka * HC + col];
        b.y = W[(ka + 1) * HC + col];
        acc = __builtin_amdgcn_wmma_f32_16x16x4_f32(false, a, false, b,
                                                    (short)0, acc, false, false);
    }
    #pragma unroll
    for (int j = 0; j < 8; ++j)
        Hbuf[(row0 + j + 8 * half) * HC + col] = acc[j];
}

// ---------------------------------------------------------------------------
// Kernel 3: a_src[n,h] = h[n,h,:]·att_src[h,:], a_dst likewise. Wave per (n,h).
// ---------------------------------------------------------------------------
__global__ __launch_bounds__(256)
void node_att_kernel(const float* __restrict__ Hbuf,
                     const float* __restrict__ att_src,
                     const float* __restrict__ att_dst,
                     float* __restrict__ a_src,
                     float* __restrict__ a_dst)
{
    int gw   = (blockIdx.x * blockDim.x + threadIdx.x) >> 5;
    int lane = threadIdx.x & 31;
    if (gw >= NNODES * NH) return;
    int n = gw >> 1, hh = gw & 1;
    const float* hrow = Hbuf + n * HC + hh * CH;
    const float* as   = att_src + hh * CH;
    const float* ad   = att_dst + hh * CH;
    float ss = 0.f, sd = 0.f;
    #pragma unroll
    for (int j = 0; j < 4; ++j) {
        int c = lane + 32 * j;
        float hv = hrow[c];
        ss += hv * as[c];
        sd += hv * ad[c];
    }
    for (int off = 16; off > 0; off >>= 1) {
        ss += __shfl_xor(ss, off, 32);
        sd += __shfl_xor(sd, off, 32);
    }
    if (lane == 0) { a_src[gw] = ss; a_dst[gw] = sd; }
}

// ---------------------------------------------------------------------------
// Kernel 4: per-edge scores + leaky ReLU + segment max. Thread per edge.
// ---------------------------------------------------------------------------
__global__ __launch_bounds__(256)
void edge_score_kernel(const int* __restrict__ ei,        // [2*E] (src row, dst row)
                       const float* __restrict__ eattr,   // [E][EDIM]
                       const float* __restrict__ Mfold,   // [EDIM*NH]
                       const float* __restrict__ c0,      // [NH]
                       const float* __restrict__ a_src,
                       const float* __restrict__ a_dst,
                       float* __restrict__ score,         // [E*NH]
                       float* __restrict__ smax)          // [N*NH]
{
    int e = blockIdx.x * blockDim.x + threadIdx.x;
    if (e >= NEDGES) return;
    int s = ei[e], d = ei[NEDGES + e];
    float ea[EDIM];
    #pragma unroll
    for (int k = 0; k < EDIM; ++k) ea[k] = eattr[e * EDIM + k];
    #pragma unroll
    for (int hh = 0; hh < NH; ++hh) {
        float ae = c0[hh];
        #pragma unroll
        for (int k = 0; k < EDIM; ++k) ae += ea[k] * Mfold[k * NH + hh];
        float sc = a_src[s * NH + hh] + a_dst[d * NH + hh] + ae;
        sc = (sc >= 0.f) ? sc : LEAKY * sc;
        score[e * NH + hh] = sc;
        atomMaxF32(&smax[d * NH + hh], sc);
    }
}

// ---------------------------------------------------------------------------
// Kernel 5: ex = exp(score - smax[dst]); denom += ex (in-place overwrite score)
// ---------------------------------------------------------------------------
__global__ __launch_bounds__(256)
void edge_exp_kernel(const int* __restrict__ ei,
                     float* __restrict__ score,           // in: score, out: ex
                     const float* __restrict__ smax,
                     float* __restrict__ denom)
{
    int e = blockIdx.x * blockDim.x + threadIdx.x;
    if (e >= NEDGES) return;
    int d = ei[NEDGES + e];
    #pragma unroll
    for (int hh = 0; hh < NH; ++hh) {
        float ex = __expf(score[e * NH + hh] - smax[d * NH + hh]);
        score[e * NH + hh] = ex;
        atomAddF32(&denom[d * NH + hh], ex);
    }
}

// ---------------------------------------------------------------------------
// Kernel 6: scatter messages. Wave per edge; head-mean (x0.5) folded in, so
// ypre is [N][DIM] (heads pre-averaged). 256 coalesced atomic f32 adds/edge (128 per head).
// ---------------------------------------------------------------------------
__global__ __launch_bounds__(256)
void edge_scatter_kernel(const int* __restrict__ ei,
                         const float* __restrict__ ex,     // [E*NH]
                         const float* __restrict__ denom,  // [N*NH]
                         const float* __restrict__ Hbuf,   // [N][HC]
                         float* __restrict__ ypre)         // [N][DIM]
{
    int gw   = (blockIdx.x * blockDim.x + threadIdx.x) >> 5;
    int lane = threadIdx.x & 31;
    if (gw >= NEDGES) return;
    int s = ei[gw], d = ei[NEDGES + gw];
    float* yrow = ypre + d * DIM;
    #pragma unroll
    for (int hh = 0; hh < NH; ++hh) {
        float alpha = ex[gw * NH + hh] / (denom[d * NH + hh] + SMEPS);
        float w = 0.5f * alpha;                       // mean over NH=2 heads
        const float* hrow = Hbuf + s * HC + hh * CH;
        #pragma unroll
        for (int j = 0; j < 4; ++j) {
            int c = lane + 32 * j;
            atomAddF32(&yrow[c], w * hrow[c]);
        }
    }
}

// ---------------------------------------------------------------------------
// Kernel 7: out = LayerNorm(ypre + bias_gat + x) * gamma + beta. Wave per node.
// ---------------------------------------------------------------------------
__global__ __launch_bounds__(256)
void finalize_kernel(const float* __restrict__ ypre,
                     const float* __restrict__ X,
                     const float* __restrict__ bias,
                     const float* __restrict__ gamma,
                     const float* __restrict__ beta,
                     float* __restrict__ out)
{
    int gw   = (blockIdx.x * blockDim.x + threadIdx.x) >> 5;
    int lane = threadIdx.x & 31;
    if (gw >= NNODES) return;
    float y[4];
    float s = 0.f, s2 = 0.f;
    #pragma unroll
    for (int j = 0; j < 4; ++j) {
        int c = lane + 32 * j;
        float v = ypre[gw * DIM + c] + bias[c] + X[gw * DIM + c];
        y[j] = v; s += v; s2 += v * v;
    }
    for (int off = 16; off > 0; off >>= 1) {
        s  += __shfl_xor(s,  off, 32);
        s2 += __shfl_xor(s2, off, 32);
    }
    const float mu  = s * (1.0f / DIM);
    const float var = s2 * (1.0f / DIM) - mu * mu;
    const float inv = rsqrtf(var + LNEPS);
    #pragma unroll
    for (int j = 0; j < 4; ++j) {
        int c = lane + 32 * j;
        out[gw * DIM + c] = (y[j] - mu) * inv * gamma[c] + beta[c];
    }
}

// ---------------------------------------------------------------------------
// Launch
// ---------------------------------------------------------------------------
static inline float* carve(char*& p, size_t nfloats) {
    float* r = (float*)p;
    size_t bytes = (nfloats * sizeof(float) + 255) & ~(size_t)255;
    p += bytes;
    return r;
}

extern "C" void kernel_launch(void* const* d_in, const int* in_sizes, int n_in,
                              void* d_out, int out_size, void* d_ws, size_t ws_size,
                              hipStream_t stream) {
    const float* x        = (const float*)d_in[0];
    const int*   ei       = (const int*)  d_in[1];   // [2][E]
    const float* eattr    = (const float*)d_in[2];
    const float* W_ep     = (const float*)d_in[3];
    const float* b_ep     = (const float*)d_in[4];
    const float* W_lin    = (const float*)d_in[5];
    const float* att_src  = (const float*)d_in[6];
    const float* att_dst  = (const float*)d_in[7];
    const float* W_le     = (const float*)d_in[8];
    const float* att_edge = (const float*)d_in[9];
    const float* bias_gat = (const float*)d_in[10];
    const float* ln_gamma = (const float*)d_in[11];
    const float* ln_beta  = (const float*)d_in[12];
    float* out = (float*)d_out;

    char* wp = (char*)d_ws;
    float* Hbuf  = carve(wp, (size_t)NNODES * HC);   // 51.2 MB
    float* ypre  = carve(wp, (size_t)NNODES * DIM);  // 25.6 MB
    float* score = carve(wp, (size_t)NEDGES * NH);   // 6.4 MB (reused as ex)
    float* a_src = carve(wp, (size_t)NNODES * NH);
    float* a_dst = carve(wp, (size_t)NNODES * NH);
    float* smax  = carve(wp, (size_t)NNODES * NH);
    float* denom = carve(wp, (size_t)NNODES * NH);
    float* Mfold = carve(wp, (size_t)EDIM * NH);
    float* c0    = carve(wp, (size_t)NH);

    prep_fold_kernel<<<1, 256, 0, stream>>>(W_le, att_edge, W_ep, b_ep, Mfold, c0);
    init_kernel<<<25000, 256, 0, stream>>>(smax, denom, ypre);
    node_proj_wmma_kernel<<<dim3(NNODES / 16, 2), 256, 0, stream>>>(x, W_lin, Hbuf);
    node_att_kernel<<<(NNODES * NH * 32 + 255) / 256, 256, 0, stream>>>(
        Hbuf, att_src, att_dst, a_src, a_dst);
    edge_score_kernel<<<(NEDGES + 255) / 256, 256, 0, stream>>>(
        ei, eattr, Mfold, c0, a_src, a_dst, score, smax);
    edge_exp_kernel<<<(NEDGES + 255) / 256, 256, 0, stream>>>(ei, score, smax, denom);
    edge_scatter_kernel<<<(NEDGES * 32 + 255) / 256, 256, 0, stream>>>(
        ei, score, denom, Hbuf, ypre);
    finalize_kernel<<<(NNODES * 32 + 255) / 256, 256, 0, stream>>>(
        ypre, x, bias_gat, ln_gamma, ln_beta, out);
}